// GraphPool_59124519796838
// MI455X (gfx1250) — compile-verified
//
#include <hip/hip_runtime.h>
#include <hip/hip_bf16.h>

typedef __attribute__((ext_vector_type(2))) float v2f;
typedef __attribute__((ext_vector_type(8))) float v8f;

#define BATCH 32
#define NNODE 4096
#define DIM   512
#define NKEEP 2048

// ---------------------------------------------------------------------------
// Stage 1: scores[b,n] = sigmoid(dot(h[b,n,:], W) + bias)
// One wave32 per 16-node tile. A = h tile (16x4 f32), B = W chunk replicated
// across 16 columns, accumulate C over K=512 in 128 WMMA_F32_16X16X4_F32 ops.
// A layout (ISA 7.12.2): VGPR0 = {K0 lanes0-15, K2 lanes16-31}, VGPR1 = {K1,K3}
// B layout (rows striped across lanes): VGPR0 = {K0,K2}, VGPR1 = {K1,K3}
// C layout: VGPR r -> M=r (lanes0-15), M=8+r (lanes16-31); all N columns equal.
// ---------------------------------------------------------------------------
__global__ __launch_bounds__(256) void score_kernel(const float* __restrict__ h,
                                                    const float* __restrict__ W,
                                                    const float* __restrict__ bias_p,
                                                    float* __restrict__ scores) {
    const int wave = (blockIdx.x << 3) + (threadIdx.x >> 5);  // 8 waves / block
    const int lane = threadIdx.x & 31;
    const int half = lane >> 4;   // 0 or 1
    const int m    = lane & 15;   // node-in-tile for A rows

    const int b        = wave >> 8;          // wave / (NNODE/16)
    const int nodeBase = (wave & 255) << 4;  // (wave % 256) * 16

    const float* __restrict__ rowA =
        h + ((size_t)b * NNODE + nodeBase + m) * DIM;

    v8f c = {};
    #pragma unroll 4
    for (int k0 = 0; k0 < DIM; k0 += 4) {
        const int ko = k0 + 2 * half;
        v2f a, bb;
        a.x  = rowA[ko];
        a.y  = rowA[ko + 1];
        bb.x = W[ko];
        bb.y = W[ko + 1];
        c = __builtin_amdgcn_wmma_f32_16x16x4_f32(
                /*neg_a=*/false, a, /*neg_b=*/false, bb,
                /*c_mod=*/(short)0, c, /*reuse_a=*/false, /*reuse_b=*/false);
    }

    // lanes 0 and 16 jointly hold all 16 node scores (8 each across C VGPRs)
    if (m == 0) {
        const float bias = bias_p[0];
        const int base = nodeBase + half * 8;
        float* dst = scores + (size_t)b * NNODE + base;
        #pragma unroll
        for (int r = 0; r < 8; ++r) {
            float x = c[r] + bias;
            dst[r] = 1.0f / (1.0f + __builtin_expf(-x));
        }
    }
}

// ---------------------------------------------------------------------------
// Stage 2: per-batch descending bitonic sort of 4096 keys in LDS.
// key = (score_bits << 32) | ~index  -> score desc, index asc tiebreak,
// matching lax.top_k / torch.topk ordering. Scores are sigmoid outputs
// (positive floats), so unsigned bit-compare == float compare.
// ---------------------------------------------------------------------------
__global__ __launch_bounds__(1024) void topk_kernel(const float* __restrict__ scores,
                                                    int* __restrict__ topIdx,
                                                    float* __restrict__ topScore) {
    __shared__ unsigned long long key[NNODE];  // 32 KB
    const int b   = blockIdx.x;
    const int tid = threadIdx.x;

    for (int i = tid; i < NNODE; i += 1024) {
        unsigned int bits = __float_as_uint(scores[(size_t)b * NNODE + i]);
        key[i] = ((unsigned long long)bits << 32) | (unsigned int)(~i);
    }
    __syncthreads();

    for (int k = 2; k <= NNODE; k <<= 1) {
        for (int j = k >> 1; j > 0; j >>= 1) {
            for (int i = tid; i < NNODE; i += 1024) {
                const int ixj = i ^ j;
                if (ixj > i) {
                    unsigned long long a  = key[i];
                    unsigned long long bb = key[ixj];
                    const bool descRegion = ((i & k) == 0);
                    const bool doSwap = descRegion ? (a < bb) : (a > bb);
                    if (doSwap) { key[i] = bb; key[ixj] = a; }
                }
            }
            __syncthreads();
        }
    }

    for (int i = tid; i < NKEEP; i += 1024) {
        unsigned long long kk = key[i];
        topIdx[b * NKEEP + i]   = (int)(~(unsigned int)kk);
        topScore[b * NKEEP + i] = __uint_as_float((unsigned int)(kk >> 32));
    }
}

// ---------------------------------------------------------------------------
// Stage 3: out[b,k,:] = h[b, topIdx[b,k], :] * topScore[b,k]
// 2 rows per 256-thread block; one float4 (128-bit) per thread per row.
// This is the bandwidth-critical stage: fully coalesced B128 loads/stores.
// ---------------------------------------------------------------------------
__global__ __launch_bounds__(256) void gather_kernel(const float* __restrict__ h,
                                                     const int* __restrict__ topIdx,
                                                     const float* __restrict__ topScore,
                                                     float* __restrict__ out) {
    const int row  = blockIdx.x * 2 + (threadIdx.x >> 7);  // [0, BATCH*NKEEP)
    const int t    = threadIdx.x & 127;                    // 128 threads * float4 = 512
    const int b    = row >> 11;                            // row / NKEEP
    const int idx  = topIdx[row];
    const float s  = topScore[row];

    const float4* __restrict__ src =
        (const float4*)(h + ((size_t)b * NNODE + idx) * DIM);
    float4* __restrict__ dst = (float4*)(out + (size_t)row * DIM);

    float4 v = src[t];
    v.x *= s; v.y *= s; v.z *= s; v.w *= s;
    dst[t] = v;
}

extern "C" void kernel_launch(void* const* d_in, const int* in_sizes, int n_in,
                              void* d_out, int out_size, void* d_ws, size_t ws_size,
                              hipStream_t stream) {
    const float* h  = (const float*)d_in[0];   // [32, 4096, 512]
    const float* W  = (const float*)d_in[1];   // [1, 512]
    const float* bp = (const float*)d_in[2];   // [1]
    float* out = (float*)d_out;                // [32, 2048, 512]

    // workspace layout
    float* scores   = (float*)d_ws;                                   // B*N floats
    int*   topIdx   = (int*)((char*)d_ws + (size_t)BATCH * NNODE * 4); // B*NKEEP ints
    float* topScore = (float*)((char*)topIdx + (size_t)BATCH * NKEEP * 4);

    // Stage 1: 8192 tile-waves, 8 waves per 256-thread block -> 1024 blocks
    score_kernel<<<dim3((BATCH * NNODE / 16) / 8), dim3(256), 0, stream>>>(
        h, W, bp, scores);

    // Stage 2: one block per batch
    topk_kernel<<<dim3(BATCH), dim3(1024), 0, stream>>>(scores, topIdx, topScore);

    // Stage 3: 2 rows per block
    gather_kernel<<<dim3((BATCH * NKEEP) / 2), dim3(256), 0, stream>>>(
        h, topIdx, topScore, out);
}